// PatchSelector1_81535659147920
// MI455X (gfx1250) — compile-verified
//
#include <hip/hip_runtime.h>

typedef __attribute__((ext_vector_type(2))) float v2f;
typedef __attribute__((ext_vector_type(8))) float v8f;
typedef __attribute__((ext_vector_type(4))) int   v4i;

#define D_MODEL 512
#define PATCH   64          // P (patches per chunk)
#define CHUNKS  32          // C
#define BATCH   32          // B
#define NPATCH  (CHUNKS * PATCH)   // 2048 patches per batch
#define SELECT_N 256
#define SORT_N  2048

#if __has_builtin(__builtin_amdgcn_global_load_async_to_lds_b128) && \
    __has_builtin(__builtin_amdgcn_s_wait_asynccnt)
#define HAVE_ASYNC_LDS 1
typedef __attribute__((address_space(1))) v4i* gptr_v4i;
typedef __attribute__((address_space(3))) v4i* lptr_v4i;
#endif

// ---------------------------------------------------------------------------
// Kernel 1: scores[b][c*64+p] = sum_d x[b,c,d,p] * w[d] + bias
// One block per (b,c) slab (512x64 f32 = 128 KiB). 4 waves, each wave owns a
// 16-patch group and runs 128 steps of V_WMMA_F32_16X16X4_F32 over K=512.
// A = 16(p) x 4(d) tile of the slab (f32 A-layout: lanes 0-15 K={0,1},
// lanes 16-31 K={2,3}); B = w[d..d+3] replicated across all 16 columns, so
// every column of D holds the 16 patch scores.
// ---------------------------------------------------------------------------
__global__ __launch_bounds__(128) void k_scores(const float* __restrict__ x,
                                                const float* __restrict__ w,
                                                const float* __restrict__ bias,
                                                float* __restrict__ scores) {
  const int blk  = blockIdx.x;        // b*32 + c
  const int tid  = threadIdx.x;
  const int wave = tid >> 5;          // 0..3 -> patch group
  const int lane = tid & 31;
  const int m     = lane & 15;        // A-matrix row (patch within group)
  const int halfk = (lane >> 4) << 1; // 0 for lanes 0-15, 2 for lanes 16-31
  const int p     = wave * 16 + m;
  const float* slab = x + (size_t)blk * (D_MODEL * PATCH);

  v8f acc = {};
  for (int d0 = 0; d0 < D_MODEL; d0 += 4) {
    const int d = d0 + halfk;
    v2f A, Bv;
    A.x  = slab[(size_t)d * PATCH + p];         // K = halfk
    A.y  = slab[(size_t)(d + 1) * PATCH + p];   // K = halfk+1
    Bv.x = w[d];
    Bv.y = w[d + 1];
    acc = __builtin_amdgcn_wmma_f32_16x16x4_f32(
        /*neg_a=*/false, A, /*neg_b=*/false, Bv,
        /*c_mod=*/(short)0, acc, /*reuse_a=*/false, /*reuse_b=*/false);
  }

  // D layout (16x16 f32): VGPR r, lanes 0-15 -> M=r ; lanes 16-31 -> M=r+8.
  // Every column identical; lanes with N==0 (lane 0 and lane 16) write out.
  if (m == 0) {
    const float bb = bias[0];
    const int baseM = (lane >> 4) * 8;
    float* sout = scores + (size_t)blk * PATCH + wave * 16 + baseM;
#pragma unroll
    for (int r = 0; r < 8; ++r) sout[r] = acc[r] + bb;
  }
}

// ---------------------------------------------------------------------------
// Kernel 2: per-batch top-256 via full bitonic sort of 2048 (score,idx)
// pairs in LDS. Descending by score, ties broken by lower index (matches
// jax.lax.top_k ordering). One block per batch, 1024 threads (2 elems each).
// ---------------------------------------------------------------------------
__global__ __launch_bounds__(1024) void k_topk(const float* __restrict__ scores,
                                               int* __restrict__ topidx) {
  __shared__ float s[SORT_N];
  __shared__ int   idx[SORT_N];
  const int b = blockIdx.x;
  const int t = threadIdx.x;
  const float* sb = scores + (size_t)b * SORT_N;
  s[t]          = sb[t];          idx[t]          = t;
  s[t + 1024]   = sb[t + 1024];   idx[t + 1024]   = t + 1024;

  for (int k = 2; k <= SORT_N; k <<= 1) {
    for (int j = k >> 1; j > 0; j >>= 1) {
      __syncthreads();
#pragma unroll
      for (int e = 0; e < 2; ++e) {
        int i = t + e * 1024;
        int ixj = i ^ j;
        if (ixj > i) {
          float si = s[i], sx = s[ixj];
          int   ii = idx[i], ix = idx[ixj];
          // "i should come first" under global descending order
          bool iFirst = (si > sx) || (si == sx && ii < ix);
          bool desc   = ((i & k) == 0);
          if (desc != iFirst) {
            s[i] = sx;   s[ixj] = si;
            idx[i] = ix; idx[ixj] = ii;
          }
        }
      }
    }
  }
  __syncthreads();
  if (t < SELECT_N) topidx[b * SELECT_N + t] = idx[t];
}

// ---------------------------------------------------------------------------
// Kernel 3: gather. One block per (b,c). Build the list of selected patches
// that live in this chunk, then stream the slab through LDS in 128x64 chunks
// and emit each selected patch as a contiguous 512-float row (coalesced
// stores). x is L2-resident (128 MiB < 192 MB), so re-reading is cheap.
// Staging uses GLOBAL_LOAD_ASYNC_TO_LDS_B128 (ASYNCcnt path); LDS row pitch
// is 68 floats (272 B, 16B-aligned rows for B128 async writes; stride mod 64
// == 4 -> only 2-way bank conflict on the column-read phase).
// ---------------------------------------------------------------------------
#define DCHUNK 128
#define LPITCH 68
__global__ __launch_bounds__(256) void k_gather(const float* __restrict__ x,
                                                const int* __restrict__ topidx,
                                                float* __restrict__ out) {
  __shared__ float tile[DCHUNK * LPITCH];
  __shared__ int selJ[SELECT_N];
  __shared__ int selP[SELECT_N];
  __shared__ int count;
  const int blk = blockIdx.x;   // b*32 + c
  const int b   = blk >> 5;
  const int c   = blk & 31;
  const int t   = threadIdx.x;

  if (t == 0) count = 0;
  __syncthreads();
  {
    int n = topidx[b * SELECT_N + t];   // blockDim == SELECT_N
    if ((n >> 6) == c) {
      int pos = atomicAdd(&count, 1);
      selJ[pos] = t;        // output row j
      selP[pos] = n & 63;   // patch column within this chunk
    }
  }
  __syncthreads();
  const int cnt = count;
  if (cnt == 0) return;

  const float* slab = x + (size_t)blk * (D_MODEL * PATCH);
  const int grp = t >> 7;   // 0..1
  const int lt  = t & 127;

  for (int d0 = 0; d0 < D_MODEL; d0 += DCHUNK) {
    __syncthreads();
    // stage 128 rows x 64 cols (16 float4 per row), coalesced
    for (int e = t; e < (DCHUNK * PATCH) / 4; e += 256) {
      int r   = e >> 4;
      int col = (e & 15) << 2;
#if defined(HAVE_ASYNC_LDS)
      __builtin_amdgcn_global_load_async_to_lds_b128(
          (gptr_v4i)(slab + (size_t)(d0 + r) * PATCH + col),
          (lptr_v4i)&tile[r * LPITCH + col],
          /*imm_offset=*/0, /*cpol=*/0);
#else
      const float4 v = *(const float4*)(slab + (size_t)(d0 + r) * PATCH + col);
      float* dst = &tile[r * LPITCH + col];
      dst[0] = v.x; dst[1] = v.y; dst[2] = v.z; dst[3] = v.w;
#endif
    }
#if defined(HAVE_ASYNC_LDS)
    __builtin_amdgcn_s_wait_asynccnt(0);
#endif
    __syncthreads();
    // each 128-thread group handles alternating selected patches
    for (int s = grp; s < cnt; s += 2) {
      out[((size_t)b * SELECT_N + selJ[s]) * D_MODEL + d0 + lt] =
          tile[lt * LPITCH + selP[s]];
    }
  }
}

// ---------------------------------------------------------------------------
extern "C" void kernel_launch(void* const* d_in, const int* in_sizes, int n_in,
                              void* d_out, int out_size, void* d_ws, size_t ws_size,
                              hipStream_t stream) {
  (void)in_sizes; (void)n_in; (void)out_size; (void)ws_size;
  const float* x    = (const float*)d_in[0];
  const float* w    = (const float*)d_in[1];
  const float* bias = (const float*)d_in[2];
  float* out = (float*)d_out;

  float* scores = (float*)d_ws;                                          // 32*2048 f32
  int*   topidx = (int*)((char*)d_ws + (size_t)BATCH * NPATCH * sizeof(float)); // 32*256 i32

  k_scores<<<dim3(BATCH * CHUNKS), dim3(128), 0, stream>>>(x, w, bias, scores);
  k_topk  <<<dim3(BATCH),          dim3(1024), 0, stream>>>(scores, topidx);
  k_gather<<<dim3(BATCH * CHUNKS), dim3(256), 0, stream>>>(x, topidx, out);
}